// InfluencerRank_74148315398323
// MI455X (gfx1250) — compile-verified
//
#include <hip/hip_runtime.h>
#include <math.h>

// Problem constants (from the reference)
#define T_  8
#define NI_ 20000
#define NK_ 50000
#define F_  128
#define H_  256
#define O_  128
#define E_  500000

#define MAXM 512          // cap on e2 edges hitting the target (expected ~25)
#define LCAP 512          // cap on e1 edges per item (expected ~10)
#define CHUNKV (256 * 4)  // edges per vectorized chunk (int4 per thread)
#define NFULL (E_ / CHUNKV)          // 488 full vector chunks
#define TAILBASE (NFULL * CHUNKV)    // 499712

typedef __attribute__((ext_vector_type(2))) float v2f;
typedef __attribute__((ext_vector_type(8))) float v8f;

// ---------------------------------------------------------------------------
// Deterministic ordered edge-match compaction, shared by both scan kernels.
// - int4 (b128) loads: 4 edges/thread/chunk, 4x fewer iterations.
// - __syncthreads_or early-out: match-free chunks (common case, matches are
//   ~25 per 500k edges) cost a single barrier and no LDS traffic.
// - Matched chunks: 4 wave32 ballots (one per vector slot) + cross-wave
//   prefix reproduce exact global edge order -> bit-deterministic replays.
// outbuf may be global (scan_e2) or LDS (agg_e1); cap limits writes.
// Returns total true match count (uncapped) via baseS.
// ---------------------------------------------------------------------------
__device__ __forceinline__ void ordered_scan(
    const int* __restrict__ dptr, const int* __restrict__ sptr, int key,
    int* outbuf, int cap, unsigned* wtot, unsigned* baseS_p) {
  const unsigned lane = threadIdx.x & 31u;
  const unsigned wid  = threadIdx.x >> 5;
  const unsigned below = (lane == 0) ? 0u : ((1u << lane) - 1u);

  // --- full vector chunks ---
  for (int base = 0; base < TAILBASE; base += CHUNKV) {
    const int e0 = base + (int)threadIdx.x * 4;
    const int4 d = *(const int4*)(dptr + e0);
    const int m0 = (d.x == key), m1 = (d.y == key);
    const int m2 = (d.z == key), m3 = (d.w == key);
    const int any = __syncthreads_or(m0 | m1 | m2 | m3);  // uniform
    if (!any) continue;

    const unsigned b0 = __builtin_amdgcn_ballot_w32(m0 != 0);
    const unsigned b1 = __builtin_amdgcn_ballot_w32(m1 != 0);
    const unsigned b2 = __builtin_amdgcn_ballot_w32(m2 != 0);
    const unsigned b3 = __builtin_amdgcn_ballot_w32(m3 != 0);
    if (lane == 0) {
      wtot[wid] = __builtin_popcount(b0) + __builtin_popcount(b1) +
                  __builtin_popcount(b2) + __builtin_popcount(b3);
    }
    __syncthreads();
    unsigned wbase = 0;
    for (unsigned w = 0; w < wid; ++w) wbase += wtot[w];
    unsigned pos = *baseS_p + wbase +
                   __builtin_popcount(b0 & below) + __builtin_popcount(b1 & below) +
                   __builtin_popcount(b2 & below) + __builtin_popcount(b3 & below);
    if (m0) { if (pos < (unsigned)cap) outbuf[pos] = sptr[e0 + 0]; ++pos; }
    if (m1) { if (pos < (unsigned)cap) outbuf[pos] = sptr[e0 + 1]; ++pos; }
    if (m2) { if (pos < (unsigned)cap) outbuf[pos] = sptr[e0 + 2]; ++pos; }
    if (m3) { if (pos < (unsigned)cap) outbuf[pos] = sptr[e0 + 3]; ++pos; }
    __syncthreads();
    if (threadIdx.x == 0) {
      unsigned tot = 0;
      for (int w = 0; w < 8; ++w) tot += wtot[w];
      *baseS_p += tot;
    }
    __syncthreads();
  }

  // --- scalar guarded tail (E mod CHUNKV = 288 edges) ---
  for (int base = TAILBASE; base < E_; base += 256) {
    const int e = base + (int)threadIdx.x;
    int match = 0, sv = 0;
    if (e < E_) {
      match = (dptr[e] == key);
      if (match) sv = sptr[e];
    }
    const int any = __syncthreads_or(match);
    if (!any) continue;
    const unsigned mask = __builtin_amdgcn_ballot_w32(match != 0);
    if (lane == 0) wtot[wid] = __builtin_popcount(mask);
    __syncthreads();
    unsigned wbase = 0;
    for (unsigned w = 0; w < wid; ++w) wbase += wtot[w];
    if (match) {
      const unsigned pos = *baseS_p + wbase + __builtin_popcount(mask & below);
      if (pos < (unsigned)cap) outbuf[pos] = sv;
    }
    __syncthreads();
    if (threadIdx.x == 0) {
      unsigned tot = 0;
      for (int w = 0; w < 8; ++w) tot += wtot[w];
      *baseS_p += tot;
    }
    __syncthreads();
  }
}

// ---------------------------------------------------------------------------
// Kernel 1: per timestep, find e2 edges with dst == target.
// Writes items[t][pos] = e2_src value (item index), cnt2[t] = true count.
// ---------------------------------------------------------------------------
__global__ void __launch_bounds__(256) scan_e2_kernel(
    const int* __restrict__ e2_dst, const int* __restrict__ e2_src,
    const int* __restrict__ target_p, int* __restrict__ cnt2,
    int* __restrict__ items) {
  const int t = blockIdx.x;
  const int target = target_p[0];
  __shared__ unsigned wtot[8];
  __shared__ unsigned baseS;
  if (threadIdx.x == 0) baseS = 0;
  __syncthreads();
  ordered_scan(e2_dst + (size_t)t * E_, e2_src + (size_t)t * E_, target,
               items + t * MAXM, MAXM, wtot, &baseS);
  if (threadIdx.x == 0) cnt2[t] = (int)baseS;
}

// ---------------------------------------------------------------------------
// Kernel 2: meanB[t][f] = mean over slots of x_item[items[t][m]][f]
// ---------------------------------------------------------------------------
__global__ void __launch_bounds__(128) mean_item_kernel(
    const float* __restrict__ x_item, const int* __restrict__ items,
    const int* __restrict__ cnt2, float* __restrict__ meanB) {
  const int t = blockIdx.x;
  const int cntT = cnt2[t];
  const int cnt = cntT < MAXM ? cntT : MAXM;
  const float denom = fmaxf((float)cntT, 1.0f);
  const int f = threadIdx.x;  // 0..127
  float acc = 0.f;
  for (int m = 0; m < cnt; ++m) {
    int it = items[t * MAXM + m];
    acc += x_item[((size_t)t * NK_ + it) * F_ + f];
  }
  meanB[t * F_ + f] = acc / denom;
}

// ---------------------------------------------------------------------------
// Kernel 3: per (t, slot m): meanA[t][m][f] = mean over e1 edges with
// d1 == items[t][m] of x_inf[s1][f]. LDS list + fixed-order summation.
// ---------------------------------------------------------------------------
__global__ void __launch_bounds__(256) agg_e1_kernel(
    const int* __restrict__ e1_dst, const int* __restrict__ e1_src,
    const float* __restrict__ x_inf, const int* __restrict__ cnt2,
    const int* __restrict__ items, float* __restrict__ meanA) {
  const int m = blockIdx.x;
  const int t = blockIdx.y;
  const int cnt = min(cnt2[t], MAXM);
  if (m >= cnt) return;  // uniform over block

  const int item = items[t * MAXM + m];
  __shared__ int list[LCAP];
  __shared__ unsigned wtot[8];
  __shared__ unsigned baseS;
  if (threadIdx.x == 0) baseS = 0;
  __syncthreads();
  ordered_scan(e1_dst + (size_t)t * E_, e1_src + (size_t)t * E_, item,
               list, LCAP, wtot, &baseS);

  const int n1 = (int)baseS;
  const int nlist = n1 < LCAP ? n1 : LCAP;
  const float denom = fmaxf((float)n1, 1.0f);
  for (int f = threadIdx.x; f < F_; f += blockDim.x) {
    float acc = 0.f;
    for (int i = 0; i < nlist; ++i) {
      acc += x_inf[((size_t)t * NI_ + list[i]) * F_ + f];
    }
    meanA[((size_t)t * MAXM + m) * F_ + f] = acc / denom;
  }
}

// ---------------------------------------------------------------------------
// Kernel 4 (WMMA): hitem[t][m][:] = relu(meanA[m] @ W1a_l.T + b1a_l
//                                        + x_item[items[m]] @ W1a_r.T)
// One wave per 16x16 output tile; K looped in steps of 4 using
// V_WMMA_F32_16X16X4_F32 (fp32 A/B/C — exact fp32 semantics).
// A layout: lane M = lane&15, VGPR pair holds K = k0+2*(lane>>4) .. +1
// B layout: lane N = lane&15, same K split;  B[k][n] = W[n][k]
// C/D layout: VGPR v -> M = v + 8*(lane>>4), N = lane&15
// ---------------------------------------------------------------------------
__global__ void __launch_bounds__(32) item_sage_wmma_kernel(
    const int* __restrict__ cnt2, const int* __restrict__ items,
    const float* __restrict__ meanA, const float* __restrict__ x_item,
    const float* __restrict__ W1a_l, const float* __restrict__ b1a_l,
    const float* __restrict__ W1a_r, float* __restrict__ hitem) {
  const int t  = blockIdx.z;
  const int mt = blockIdx.y;
  const int n0 = blockIdx.x * 16;
  const int cnt = min(cnt2[t], MAXM);
  if (mt * 16 >= cnt) return;  // uniform: EXEC stays all-ones for WMMA

  const unsigned lane = threadIdx.x;        // 0..31, full wave
  const unsigned row  = lane & 15u;         // M (for A) or N (for B)
  const unsigned hi   = lane >> 4;          // K-half select

  const int m = mt * 16 + (int)row;
  const int it = (m < cnt) ? items[t * MAXM + m] : 0;

  const float* aptr  = meanA + ((size_t)t * MAXM + m) * F_;
  const float* xptr  = x_item + ((size_t)t * NK_ + it) * F_;
  const float* blptr = W1a_l + (size_t)(n0 + row) * F_;
  const float* brptr = W1a_r + (size_t)(n0 + row) * F_;
  const float  bias  = b1a_l[n0 + row];

  v8f acc;
  for (int v = 0; v < 8; ++v) acc[v] = bias;

  for (int k0 = 0; k0 < F_; k0 += 4) {
    const int kk = k0 + 2 * (int)hi;
    v2f a1; a1.x = aptr[kk];  a1.y = aptr[kk + 1];
    v2f b1; b1.x = blptr[kk]; b1.y = blptr[kk + 1];
    acc = __builtin_amdgcn_wmma_f32_16x16x4_f32(false, a1, false, b1,
                                                (short)0, acc, false, false);
    v2f a2; a2.x = xptr[kk];  a2.y = xptr[kk + 1];
    v2f b2; b2.x = brptr[kk]; b2.y = brptr[kk + 1];
    acc = __builtin_amdgcn_wmma_f32_16x16x4_f32(false, a2, false, b2,
                                                (short)0, acc, false, false);
  }

  for (int v = 0; v < 8; ++v) {
    const int mm = mt * 16 + v + 8 * (int)hi;
    if (mm < cnt) {
      float val = acc[v];
      val = fmaxf(val, 0.f);  // relu
      hitem[((size_t)t * MAXM + mm) * H_ + (n0 + row)] = val;
    }
  }
}

// ---------------------------------------------------------------------------
// Kernel 5: hinf[t][j] = relu(W1b_l[j]·meanB[t] + b1b_l[j]
//                             + W1b_r[j]·x_inf[t][target])
// ---------------------------------------------------------------------------
__global__ void __launch_bounds__(256) hinf_kernel(
    const float* __restrict__ meanB, const float* __restrict__ x_inf,
    const int* __restrict__ target_p, const float* __restrict__ W1b_l,
    const float* __restrict__ b1b_l, const float* __restrict__ W1b_r,
    float* __restrict__ hinf) {
  const int t = blockIdx.x;
  const int j = threadIdx.x;  // 0..255
  const int target = target_p[0];
  const float* mb = meanB + t * F_;
  const float* xr = x_inf + ((size_t)t * NI_ + target) * F_;
  float acc = b1b_l[j];
  for (int k = 0; k < F_; ++k) {
    acc += W1b_l[j * F_ + k] * mb[k] + W1b_r[j * F_ + k] * xr[k];
  }
  hinf[t * H_ + j] = fmaxf(acc, 0.f);
}

// ---------------------------------------------------------------------------
// Kernel 6: seq[t][o] = (mean over slots of hitem) @ W2b_l.T + b2b_l
//                       + hinf[t] @ W2b_r.T
// ---------------------------------------------------------------------------
__global__ void __launch_bounds__(256) oinf_kernel(
    const int* __restrict__ cnt2, const float* __restrict__ hitem,
    const float* __restrict__ hinf, const float* __restrict__ W2b_l,
    const float* __restrict__ b2b_l, const float* __restrict__ W2b_r,
    float* __restrict__ seq) {
  const int t = blockIdx.x;
  const int j = threadIdx.x;  // 0..255
  __shared__ float meanH[H_];
  __shared__ float hs[H_];
  const int cntT = cnt2[t];
  const int cnt = cntT < MAXM ? cntT : MAXM;
  const float denom = fmaxf((float)cntT, 1.0f);
  float s = 0.f;
  for (int m = 0; m < cnt; ++m) s += hitem[((size_t)t * MAXM + m) * H_ + j];
  meanH[j] = s / denom;
  hs[j] = hinf[t * H_ + j];
  __syncthreads();
  if (j < O_) {
    float acc = b2b_l[j];
    for (int k = 0; k < H_; ++k) {
      acc += W2b_l[j * H_ + k] * meanH[k] + W2b_r[j * H_ + k] * hs[k];
    }
    seq[t * O_ + j] = acc;
  }
}

// ---------------------------------------------------------------------------
// Kernel 7: GRU over T steps + softmax attention + prediction head -> scalar
// Single block, 256 threads; all reductions in fixed order (deterministic).
// ---------------------------------------------------------------------------
__global__ void __launch_bounds__(256) head_kernel(
    const float* __restrict__ seq,
    const float* __restrict__ W_ih, const float* __restrict__ W_hh,
    const float* __restrict__ b_ih, const float* __restrict__ b_hh,
    const float* __restrict__ W_att, const float* __restrict__ b_att,
    const float* __restrict__ W_p1, const float* __restrict__ b_p1,
    const float* __restrict__ W_p2, const float* __restrict__ b_p2,
    float* __restrict__ out) {
  __shared__ float h[H_];
  __shared__ float rnn[T_][H_];
  __shared__ float xsh[O_];
  __shared__ float logit[T_];
  __shared__ float attw[T_];
  __shared__ float ctx[H_];
  __shared__ float hid[O_];
  __shared__ float red[O_];

  const int j = threadIdx.x;  // 0..255
  h[j] = 0.f;
  __syncthreads();

  for (int t = 0; t < T_; ++t) {
    if (j < O_) xsh[j] = seq[t * O_ + j];
    __syncthreads();
    // input gates (gate order r, z, n)
    float gir = b_ih[j], giz = b_ih[H_ + j], gin = b_ih[2 * H_ + j];
    for (int k = 0; k < O_; ++k) {
      float x = xsh[k];
      gir += W_ih[(size_t)j * O_ + k] * x;
      giz += W_ih[(size_t)(H_ + j) * O_ + k] * x;
      gin += W_ih[(size_t)(2 * H_ + j) * O_ + k] * x;
    }
    float ghr = b_hh[j], ghz = b_hh[H_ + j], ghn = b_hh[2 * H_ + j];
    for (int k = 0; k < H_; ++k) {
      float hv = h[k];
      ghr += W_hh[(size_t)j * H_ + k] * hv;
      ghz += W_hh[(size_t)(H_ + j) * H_ + k] * hv;
      ghn += W_hh[(size_t)(2 * H_ + j) * H_ + k] * hv;
    }
    float r = 1.f / (1.f + expf(-(gir + ghr)));
    float z = 1.f / (1.f + expf(-(giz + ghz)));
    float n = tanhf(gin + r * ghn);
    float hnew = (1.f - z) * n + z * h[j];
    __syncthreads();           // all reads of h done
    h[j] = hnew;
    rnn[t][j] = hnew;
    __syncthreads();
  }

  // attention logits (threads 0..7, fixed-order dots)
  if (j < T_) {
    float s = b_att[0];
    for (int k = 0; k < H_; ++k) s += W_att[k] * rnn[j][k];
    logit[j] = s;
  }
  __syncthreads();
  if (j == 0) {
    float mx = logit[0];
    for (int t = 1; t < T_; ++t) mx = fmaxf(mx, logit[t]);
    float ss = 0.f;
    for (int t = 0; t < T_; ++t) { attw[t] = expf(logit[t] - mx); ss += attw[t]; }
    for (int t = 0; t < T_; ++t) attw[t] /= ss;
  }
  __syncthreads();
  float c = 0.f;
  for (int t = 0; t < T_; ++t) c += attw[t] * rnn[t][j];
  ctx[j] = c;
  __syncthreads();
  if (j < O_) {
    float a = b_p1[j];
    for (int k = 0; k < H_; ++k) a += W_p1[(size_t)j * H_ + k] * ctx[k];
    hid[j] = fmaxf(a, 0.f);
  }
  __syncthreads();
  if (j < O_) red[j] = hid[j] * W_p2[j];
  __syncthreads();
  for (int s = O_ / 2; s > 0; s >>= 1) {
    if (j < s) red[j] += red[j + s];
    __syncthreads();
  }
  if (j == 0) out[0] = red[0] + b_p2[0];
}

// ---------------------------------------------------------------------------
extern "C" void kernel_launch(void* const* d_in, const int* in_sizes, int n_in,
                              void* d_out, int out_size, void* d_ws, size_t ws_size,
                              hipStream_t stream) {
  const float* x_inf  = (const float*)d_in[0];
  const float* x_item = (const float*)d_in[1];
  const int*   e1_src = (const int*)d_in[2];
  const int*   e1_dst = (const int*)d_in[3];
  const int*   e2_src = (const int*)d_in[4];
  const int*   e2_dst = (const int*)d_in[5];
  const int*   target = (const int*)d_in[6];
  const float* W1a_l  = (const float*)d_in[7];
  const float* b1a_l  = (const float*)d_in[8];
  const float* W1a_r  = (const float*)d_in[9];
  const float* W1b_l  = (const float*)d_in[10];
  const float* b1b_l  = (const float*)d_in[11];
  const float* W1b_r  = (const float*)d_in[12];
  // d_in[13..15] (W2a_l, b2a_l, W2a_r) are dead code in the reference
  const float* W2b_l  = (const float*)d_in[16];
  const float* b2b_l  = (const float*)d_in[17];
  const float* W2b_r  = (const float*)d_in[18];
  const float* W_ih   = (const float*)d_in[19];
  const float* W_hh   = (const float*)d_in[20];
  const float* b_ih   = (const float*)d_in[21];
  const float* b_hh   = (const float*)d_in[22];
  const float* W_att  = (const float*)d_in[23];
  const float* b_att  = (const float*)d_in[24];
  const float* W_p1   = (const float*)d_in[25];
  const float* b_p1   = (const float*)d_in[26];
  const float* W_p2   = (const float*)d_in[27];
  const float* b_p2   = (const float*)d_in[28];

  // Workspace carve-up (~6.3 MB total; every live byte rewritten each call)
  char* ws = (char*)d_ws;
  size_t off = 0;
  int*   d_items = (int*)(ws + off);   off += (size_t)T_ * MAXM * sizeof(int);
  int*   d_cnt2  = (int*)(ws + off);   off += (size_t)T_ * sizeof(int);
  float* d_meanB = (float*)(ws + off); off += (size_t)T_ * F_ * sizeof(float);
  float* d_meanA = (float*)(ws + off); off += (size_t)T_ * MAXM * F_ * sizeof(float);
  float* d_hitem = (float*)(ws + off); off += (size_t)T_ * MAXM * H_ * sizeof(float);
  float* d_hinf  = (float*)(ws + off); off += (size_t)T_ * H_ * sizeof(float);
  float* d_seq   = (float*)(ws + off); off += (size_t)T_ * O_ * sizeof(float);
  (void)ws_size; (void)in_sizes; (void)n_in; (void)out_size;

  scan_e2_kernel<<<T_, 256, 0, stream>>>(e2_dst, e2_src, target, d_cnt2, d_items);
  mean_item_kernel<<<T_, 128, 0, stream>>>(x_item, d_items, d_cnt2, d_meanB);
  agg_e1_kernel<<<dim3(MAXM, T_), 256, 0, stream>>>(e1_dst, e1_src, x_inf,
                                                    d_cnt2, d_items, d_meanA);
  item_sage_wmma_kernel<<<dim3(H_ / 16, MAXM / 16, T_), 32, 0, stream>>>(
      d_cnt2, d_items, d_meanA, x_item, W1a_l, b1a_l, W1a_r, d_hitem);
  hinf_kernel<<<T_, 256, 0, stream>>>(d_meanB, x_inf, target, W1b_l, b1b_l,
                                      W1b_r, d_hinf);
  oinf_kernel<<<T_, 256, 0, stream>>>(d_cnt2, d_hitem, d_hinf, W2b_l, b2b_l,
                                      W2b_r, d_seq);
  head_kernel<<<1, 256, 0, stream>>>(d_seq, W_ih, W_hh, b_ih, b_hh, W_att,
                                     b_att, W_p1, b_p1, W_p2, b_p2,
                                     (float*)d_out);
}